// path_generator_GAD_33603824123974
// MI455X (gfx1250) — compile-verified
//
#include <hip/hip_runtime.h>
#include <stdint.h>

// SDE path generator: s[t] = s + (b0 + b1*s)*dt + (a0 + a1*max(s,0))^g * z*sqrt(dt)
// HBM-bound: ~920 MB traffic -> ~40 us floor at 23.3 TB/s. One thread per path.
// CDNA5 path: double-buffered global->LDS async staging (ASYNCcnt), shared
// INST_OFFSET for LDS+global addressing, native v_log_f32/v_exp_f32 pow,
// non-temporal output stores. 60 KB LDS/WG -> 5 workgroups per WGP.

#define PATH_T 251
#define CHUNK  5           // timesteps per stage (LDS i-stride = 4B = global stride)
#define NCHUNK 50          // 50 * 5 = 250 steps
#define BLOCK  256

// Low 32 bits of a generic pointer into LDS == LDS byte offset (aperture layout).
__device__ __forceinline__ uint32_t lds_lo32(const void* p) {
    return (uint32_t)(uintptr_t)p;
}

// Async DMA one dword per lane. INST_OFFSET is added to BOTH the LDS address
// and the global address, so one LDS base VGPR +
// one global base pointer cover the whole chunk with literal offsets.
#define ASYNC_OFF(ldsb, gp, K)                                              \
    asm volatile("global_load_async_to_lds_b32 %0, %1, off offset:" #K     \
                 :: "v"(ldsb), "v"(gp) : "memory")

#define ISSUE_ARRAY(ldsb, gp)                                               \
    do {                                                                    \
        ASYNC_OFF(ldsb, gp, 0);                                             \
        ASYNC_OFF(ldsb, gp, 4);                                             \
        ASYNC_OFF(ldsb, gp, 8);                                             \
        ASYNC_OFF(ldsb, gp, 12);                                            \
        ASYNC_OFF(ldsb, gp, 16);                                            \
    } while (0)

__device__ __forceinline__ void wait_async_le30(void) {
    // <=30 outstanding: the previous chunk's 30 copies are done (in-order done).
    asm volatile("s_wait_asynccnt 30" ::: "memory");
}
__device__ __forceinline__ void wait_async_0(void) {
    asm volatile("s_wait_asynccnt 0" ::: "memory");
}

// smem layout: [stage][array][lane][i] -> per-i stride 4B; lane stride = 5
// dwords (coprime with 32) -> conflict-free ds_load_b32 across a wave.
#define SBASE(stage, a, tid) ((((stage)*6 + (a))*BLOCK + (tid))*CHUNK)

__launch_bounds__(BLOCK)
__global__ void sde_cev_paths_kernel(
    const float* __restrict__ Z,
    const float* __restrict__ u_a0,
    const float* __restrict__ u_a1,
    const float* __restrict__ u_b0,
    const float* __restrict__ u_b1,
    const float* __restrict__ u_g,
    float* __restrict__ out,
    int n_paths)
{
    __shared__ float smem[2 * 6 * BLOCK * CHUNK];   // 61440 B, double-buffered

    const int tid = threadIdx.x;
    const int n   = blockIdx.x * BLOCK + tid;
    if (n >= n_paths) return;   // N is a multiple of BLOCK; no barriers used

    const size_t row = (size_t)n * PATH_T;

    const float dt   = 0.004f;
    const float sqdt = 0.06324555320336759f;        // sqrt(0.004)

    // Global stream cursors, advanced by CHUNK floats after each issue.
    const float* qz = Z    + row + 1;
    const float* q0 = u_a0 + row + 1;
    const float* q1 = u_a1 + row + 1;
    const float* q2 = u_b0 + row + 1;
    const float* q3 = u_b1 + row + 1;
    const float* q4 = u_g  + row + 1;
    float*       po = out  + row;

    // LDS destination bases (per stage, per array), computed once.
    uint32_t lb[2][6];
#pragma unroll
    for (int st = 0; st < 2; ++st)
#pragma unroll
        for (int a = 0; a < 6; ++a)
            lb[st][a] = lds_lo32(&smem[SBASE(st, a, tid)]);

    float s = 100.0f;                               // S0
    __builtin_nontemporal_store(s, po);             // out[:,0] = S0

    // Prime the pipeline: stage 0 <- chunk 0 (30 async dword copies per lane).
    ISSUE_ARRAY(lb[0][0], qz);  qz += CHUNK;
    ISSUE_ARRAY(lb[0][1], q0);  q0 += CHUNK;
    ISSUE_ARRAY(lb[0][2], q1);  q1 += CHUNK;
    ISSUE_ARRAY(lb[0][3], q2);  q2 += CHUNK;
    ISSUE_ARRAY(lb[0][4], q3);  q3 += CHUNK;
    ISSUE_ARRAY(lb[0][5], q4);  q4 += CHUNK;

    for (int c = 0; c < NCHUNK; ++c) {
        const int stage = c & 1;
        if (c + 1 < NCHUNK) {
            const int nst = stage ^ 1;
            ISSUE_ARRAY(lb[nst][0], qz);  qz += CHUNK;
            ISSUE_ARRAY(lb[nst][1], q0);  q0 += CHUNK;
            ISSUE_ARRAY(lb[nst][2], q1);  q1 += CHUNK;
            ISSUE_ARRAY(lb[nst][3], q2);  q2 += CHUNK;
            ISSUE_ARRAY(lb[nst][4], q3);  q3 += CHUNK;
            ISSUE_ARRAY(lb[nst][5], q4);  q4 += CHUNK;
            wait_async_le30();          // current stage fully landed in LDS
        } else {
            wait_async_0();
        }

        const int tb = 1 + c * CHUNK;
#pragma unroll
        for (int i = 0; i < CHUNK; ++i) {
            const float zt  = smem[SBASE(stage, 0, tid) + i] * sqdt;
            const float ua0 = smem[SBASE(stage, 1, tid) + i];
            const float ua1 = smem[SBASE(stage, 2, tid) + i];
            const float ub0 = smem[SBASE(stage, 3, tid) + i];
            const float ub1 = smem[SBASE(stage, 4, tid) + i];
            const float ug  = smem[SBASE(stage, 5, tid) + i];

            // Affine coefficient maps folded into FMAs.
            const float a0 = fmaf(0.10f, ua0, 0.05f);   // [0.05, 0.15]
            const float a1 = fmaf(0.20f, ua1, 0.10f);   // [0.10, 0.30]
            const float b0 = 0.05f * ub0;               // [0.00, 0.05]
            const float b1 = 0.10f * ub1;               // [0.00, 0.10]
            const float g  = fmaf(0.20f, ug,  0.80f);   // [0.80, 1.00]

            // base >= 0.05 always -> native pow: exp2(g * log2(base))
            const float base = fmaf(a1, fmaxf(s, 0.0f), a0);
            const float vol  = __builtin_amdgcn_exp2f(g * __builtin_amdgcn_logf(base));

            // s += drift*dt + vol*zt
            s = fmaf(fmaf(b1, s, b0), dt, fmaf(vol, zt, s));

            __builtin_nontemporal_store(s, po + tb + i);
        }
    }
}

extern "C" void kernel_launch(void* const* d_in, const int* in_sizes, int n_in,
                              void* d_out, int out_size, void* d_ws, size_t ws_size,
                              hipStream_t stream) {
    const float* Z    = (const float*)d_in[0];
    const float* u_a0 = (const float*)d_in[1];
    const float* u_a1 = (const float*)d_in[2];
    const float* u_b0 = (const float*)d_in[3];
    const float* u_b1 = (const float*)d_in[4];
    const float* u_g  = (const float*)d_in[5];
    // d_in[6] is sample_size (device scalar) -- derive N on host instead:
    const int n_paths = in_sizes[0] / PATH_T;   // in_sizes[0] == N*T

    float* out = (float*)d_out;

    dim3 block(BLOCK);
    dim3 grid((n_paths + BLOCK - 1) / BLOCK);
    sde_cev_paths_kernel<<<grid, block, 0, stream>>>(
        Z, u_a0, u_a1, u_b0, u_b1, u_g, out, n_paths);
}